// kmeans_67181878444556
// MI455X (gfx1250) — compile-verified
//
#include <hip/hip_runtime.h>

typedef __attribute__((ext_vector_type(2))) float v2f;
typedef __attribute__((ext_vector_type(8))) float v8f;

#define N_TOTAL 9600
#define D_DIM 128
#define K_DIM 16
#define NB 192                       // n rows per block
#define NBLOCKS (N_TOTAL / NB)       // 50

// d_ws layout (floats): [0 .. D*K-1] = M accumulator (row-major D x K),
//                       [D*K]        = tr(h h^T) accumulator
__global__ __launch_bounds__(256) void gram_wmma_kernel(const float* __restrict__ h,
                                                        const float* __restrict__ F,
                                                        float* __restrict__ ws) {
  const int lane = threadIdx.x & 31;
  const int wave = threadIdx.x >> 5;   // 0..7 -> d-tile index
  const int half = lane >> 4;          // 0: lanes 0-15, 1: lanes 16-31
  const int l16  = lane & 15;
  const int dcol = wave * 16 + l16;    // d index served by this lane
  const int n0base = blockIdx.x * NB;

  v8f c = {};          // 16x16 f32 accumulator (partial M tile)
  float ssq = 0.f;     // partial sum(h*h)

  for (int n0 = n0base; n0 < n0base + NB; n0 += 4) {
    // A = 16(d) x 4(n) tile of h^T, f32 layout (ISA 7.12.2):
    //   lanes 0-15 hold K=0,1 (n0+0,n0+1); lanes 16-31 hold K=2,3 (n0+2,n0+3)
    const int nA = n0 + half * 2;
    const float a0 = h[(size_t)nA * D_DIM + dcol];
    const float a1 = h[(size_t)(nA + 1) * D_DIM + dcol];
    v2f a; a.x = a0; a.y = a1;

    // B = 4(n) x 16(k) tile of F:
    //   VGPR0: rows K=0 (lanes 0-15) / K=2 (lanes 16-31); VGPR1: K=1 / K=3
    const float b0 = F[(size_t)nA * K_DIM + l16];
    const float b1 = F[(size_t)(nA + 1) * K_DIM + l16];
    v2f b; b.x = b0; b.y = b1;

    ssq += a0 * a0 + a1 * a1;   // each h element touched exactly once grid-wide

    // v_wmma_f32_16x16x4_f32: D = A x B + C
    c = __builtin_amdgcn_wmma_f32_16x16x4_f32(false, a, false, b, (short)0, c,
                                              false, false);
  }

  // Spill partial M tile: C VGPR r -> row (r + 8*half) of this d-tile, col l16
  #pragma unroll
  for (int r = 0; r < 8; ++r) {
    const int drow = wave * 16 + r + 8 * half;
    atomicAdd(&ws[drow * K_DIM + l16], c[r]);
  }

  // Wave-reduce the trace partial, one atomic per wave (wave32)
  for (int off = 16; off > 0; off >>= 1)
    ssq += __shfl_down(ssq, off, 32);
  if (lane == 0) atomicAdd(&ws[D_DIM * K_DIM], ssq);
}

__global__ __launch_bounds__(256) void finalize_kernel(const float* __restrict__ ws,
                                                       float* __restrict__ out) {
  __shared__ float red[8];
  float s = 0.f;
  for (int i = threadIdx.x; i < D_DIM * K_DIM; i += 256) {
    const float m = ws[i];
    s += m * m;
  }
  for (int off = 16; off > 0; off >>= 1)
    s += __shfl_down(s, off, 32);
  if ((threadIdx.x & 31) == 0) red[threadIdx.x >> 5] = s;
  __syncthreads();
  if (threadIdx.x == 0) {
    float t = 0.f;
    #pragma unroll
    for (int i = 0; i < 8; ++i) t += red[i];
    out[0] = ws[D_DIM * K_DIM] - t;   // tr(hh^T) - ||h^T F||_F^2
  }
}

extern "C" void kernel_launch(void* const* d_in, const int* in_sizes, int n_in,
                              void* d_out, int out_size, void* d_ws, size_t ws_size,
                              hipStream_t stream) {
  const float* h = (const float*)d_in[0];   // [9600 x 128] f32
  const float* F = (const float*)d_in[1];   // [9600 x 16]  f32
  float* ws = (float*)d_ws;

  // Zero the M + trace accumulators (graph-capture-safe async memset)
  hipMemsetAsync(d_ws, 0, (D_DIM * K_DIM + 1) * sizeof(float), stream);

  gram_wmma_kernel<<<NBLOCKS, 256, 0, stream>>>(h, F, ws);
  finalize_kernel<<<1, 256, 0, stream>>>(ws, (float*)d_out);
}